// AdaptiveLinearWithChannel_11982958756449
// MI455X (gfx1250) — compile-verified
//
#include <hip/hip_runtime.h>

// Problem constants (from reference): C=1024, IN=256, OUT=256, B=256, N=1024
#define NROWS   1024
#define KDIM    256
#define ODIM    256
#define WAVES   16
#define THREADS (WAVES * 32)
#define KCHUNK  32                      // K rows per stage
#define NSTAGES (KDIM / KCHUNK)         // 8
#define STAGE_FLOATS (KCHUNK * ODIM)    // 8192 floats = 32 KB per buffer
#define KSTEPS  (KCHUNK / 4)            // 8 WMMA K-steps per stage

typedef __attribute__((ext_vector_type(2))) float v2f;
typedef __attribute__((ext_vector_type(4))) float v4f;
typedef __attribute__((ext_vector_type(8))) float v8f;

typedef __attribute__((address_space(1))) int* g_i32_ptr;   // device-global int*
typedef __attribute__((address_space(3))) int* lds_i32_ptr; // LDS int*

#if defined(__has_builtin)
#if __has_builtin(__builtin_amdgcn_global_load_async_to_lds_b32) && \
    __has_builtin(__builtin_amdgcn_s_wait_asynccnt)
#define USE_ASYNC_LDS 1
#endif
#endif

#ifdef USE_ASYNC_LDS
#define ASYNC_FENCE() __builtin_amdgcn_s_wait_asynccnt(0)
#else
#define ASYNC_FENCE()
#endif

// Stage W rows [ks*32, ks*32+32) x 256 cols into LDS buffer `buf`, in
// pair-interleaved, odd-pair-XOR16-swizzled layout:
//   lds[(k>>1)*512 + 2*(c ^ (((k>>1)&1)<<4)) + (k&1)] = W[k][c]
__device__ __forceinline__ void stage_load(const float* __restrict__ wt,
                                           float* __restrict__ wlds,
                                           int tid, int ks, int buf)
{
    const float* __restrict__ src = wt + ks * STAGE_FLOATS;
    float* __restrict__ dst = wlds + buf * STAGE_FLOATS;
#ifdef USE_ASYNC_LDS
    // Per-thread: cc = tid&255 (const), k = (tid>>8) + 2*it  -> k>>1 = it,
    // k&1 = tid>>8 (const). Global side stays lane-consecutive (coalesced);
    // LDS side scatters into the swizzled layout.
    const int cc   = tid & 255;
    const int k0   = tid >> 8;                 // 0 or 1
    const int offE = 2 * cc + k0;              // even it: swizzle = 0
    const int offO = 2 * (cc ^ 16) + k0;       // odd  it: swizzle = 16
    #pragma unroll
    for (int it = 0; it < STAGE_FLOATS / THREADS; ++it) {   // 16 iters
        const int off = (it << 9) + ((it & 1) ? offO : offE);
        __builtin_amdgcn_global_load_async_to_lds_b32(
            (g_i32_ptr)(src + tid + (it << 9)),
            (lds_i32_ptr)(dst + off),
            0, 0);
    }
#else
    #pragma unroll
    for (int it = 0; it < STAGE_FLOATS / (4 * THREADS); ++it) { // 4 iters
        const int i4 = tid + it * THREADS;
        const int k  = i4 >> 6;            // row within chunk
        const int c0 = (i4 & 63) << 2;     // column base
        const v4f v  = *(const v4f*)(src + (k << 8) + c0);
        const int sw = ((k >> 1) & 1) << 4;
        float* d = dst + ((k >> 1) << 9) + (k & 1) + ((c0 ^ sw) << 1);
        #pragma unroll
        for (int j = 0; j < 4; ++j) d[j << 1] = v[j];
    }
#endif
}

__global__ __launch_bounds__(THREADS) void alwc_kernel(
    const float* __restrict__ x,       // [B, N, IN]
    const int*   __restrict__ indices, // [B]
    const float* __restrict__ weight,  // [C, IN, OUT]
    const float* __restrict__ bias,    // [C, 1, OUT]
    float* __restrict__ out)           // [B, N, OUT]
{
    __shared__ float wlds[2 * STAGE_FLOATS]; // 64 KB double buffer

    const int b  = blockIdx.x;
    const int ch = indices[b];
    const float* __restrict__ wt = weight + (size_t)ch * (KDIM * ODIM);
    const float* __restrict__ bs = bias   + (size_t)ch * ODIM;
    const float* __restrict__ xb = x      + (size_t)b  * (NROWS * KDIM);
    float*       __restrict__ ob = out    + (size_t)b  * (NROWS * ODIM);

    const int tid   = threadIdx.x;
    const int wave  = tid >> 5;
    const int lane  = tid & 31;
    const int lhalf = lane >> 4;   // 0: lanes 0-15, 1: lanes 16-31
    const int lmod  = lane & 15;

    const v8f vzero = {0.f, 0.f, 0.f, 0.f, 0.f, 0.f, 0.f, 0.f};

    // 4 M-passes: 16 waves x 16-row tiles = 256 rows per pass.
    for (int mp = 0; mp < 4; ++mp) {
        const int mtile = mp * WAVES + wave;              // [0, 64)
        const int arow  = mtile * 16 + lmod;              // this lane's A row
        const float* __restrict__ xrow = xb + (size_t)arow * KDIM;

        v8f c[16];
        #pragma unroll
        for (int n = 0; n < 16; ++n) c[n] = vzero;

        // Prologue: stage chunk 0 into buffer 0.
        stage_load(wt, wlds, tid, 0, 0);
        ASYNC_FENCE();
        __syncthreads();

        for (int ks = 0; ks < NSTAGES; ++ks) {
            const int cur = ks & 1;
            // Prefetch next chunk into the other buffer (overlaps compute).
            if (ks + 1 < NSTAGES) stage_load(wt, wlds, tid, ks + 1, cur ^ 1);

            const float* __restrict__ lbase = wlds + cur * STAGE_FLOATS;

            // A fragments: lane holds x[arow][k0], x[arow][k0+1],
            // k0 = ks*32 + s*4 + 2*lhalf  (single b64 load each)
            v2f a[KSTEPS];
            const float* __restrict__ xk = xrow + ks * KCHUNK + lhalf * 2;
            #pragma unroll
            for (int s = 0; s < KSTEPS; ++s) {
                a[s] = *(const v2f*)(xk + s * 4);
            }

            // 16 N-tiles x 8 K-steps; each B fragment is one aligned
            // ds_load_b64 at base + s*4096B (immediate offset).
            #pragma unroll
            for (int n = 0; n < 16; ++n) {
                const float* __restrict__ bptr =
                    lbase + lhalf * 512 + ((((n ^ lhalf) << 4) + lmod) << 1);
                #pragma unroll
                for (int s = 0; s < KSTEPS; ++s) {
                    const v2f bf = *(const v2f*)(bptr + (s << 10));
                    c[n] = __builtin_amdgcn_wmma_f32_16x16x4_f32(
                        false, a[s], false, bf, (short)0, c[n], false, false);
                }
            }

            ASYNC_FENCE();      // prefetch complete (ran during compute)
            __syncthreads();    // LDS visible / safe to overwrite
        }

        // Bias + nontemporal store.
        // C layout: VGPR r -> row (mtile*16 + lhalf*8 + r), col (n*16 + lmod)
        #pragma unroll
        for (int n = 0; n < 16; ++n) {
            const int ncol = n * 16 + lmod;
            const float bv = bs[ncol];
            float* __restrict__ orow =
                ob + (size_t)(mtile * 16 + lhalf * 8) * ODIM + ncol;
            #pragma unroll
            for (int r = 0; r < 8; ++r) {
                __builtin_nontemporal_store(c[n][r] + bv, orow + (size_t)r * ODIM);
            }
        }
    }
}

extern "C" void kernel_launch(void* const* d_in, const int* in_sizes, int n_in,
                              void* d_out, int out_size, void* d_ws, size_t ws_size,
                              hipStream_t stream) {
    const float* x       = (const float*)d_in[0];
    const int*   indices = (const int*)d_in[1];
    const float* weight  = (const float*)d_in[2];
    const float* bias    = (const float*)d_in[3];
    float*       out     = (float*)d_out;

    const int B = in_sizes[1]; // one workgroup per batch
    dim3 grid(B), block(THREADS);
    hipLaunchKernelGGL(alwc_kernel, grid, block, 0, stream,
                       x, indices, weight, bias, out);
}